// PCF_9165460209717
// MI455X (gfx1250) — compile-verified
//
#include <hip/hip_runtime.h>
#include <hip/hip_bf16.h>

typedef float v2f __attribute__((ext_vector_type(2)));
typedef float v4f __attribute__((ext_vector_type(4)));
typedef float v8f __attribute__((ext_vector_type(8)));

// Problem shape constants from the reference (B=2, C=64, K=16, H=8, C_mid=16).
#define B_DIM   2
#define C_DIM   64
#define K_DIM   16
#define H_DIM   8
#define W_DIM   16  // C_mid

// One wave32 per point. out[c,w] = sum_k (G[k,c]*guid[k,c/8]) * W[k,w], a
// 64x16x16 f32 GEMM per point = four 16x16 WMMA tiles x four K=4 steps of
// V_WMMA_F32_16X16X4_F32.
//
// Channel permutation: lane l15 (A-matrix row) owns channels
//     c(t) = 8*(l15>>1) + 4*(l15&1) + t        (t = tile index 0..3)
// so (a) the guidance head h = c>>3 = l15>>1 is constant per lane -> guidance
// loaded once per K-slot (8 loads, reused by all tiles), and (b) the four
// tile-channels are contiguous -> each gathered feature row is ONE b128 load
// whose half-wave footprint tiles the full 256B row (perfect coalescing).
__global__ __launch_bounds__(256)
void pcf_wmma_f32_kernel(const float* __restrict__ feat,        // [B, N, 64]
                         const long long* __restrict__ inds,    // [B, M, 16] int64
                         const float* __restrict__ guid,        // [B, M, 16, 8]
                         const float* __restrict__ wn,          // [B, M, 16, 16]
                         float* __restrict__ out,               // [B, M, 1024]
                         int N, int M, int P)                   // P = B*M, B = 2
{
    const int point = blockIdx.x * (blockDim.x >> 5) + (threadIdx.x >> 5);
    if (point >= P) return;                 // uniform per wave -> EXEC all-1s for WMMA

    const int lane  = threadIdx.x & 31;
    const int half  = lane >> 4;            // selects K slots {k0,k0+1} vs {k0+2,k0+3}
    const int l15   = lane & 15;
    const int hsel  = l15 >> 1;             // guidance head for this lane (all tiles)
    const int cbase = 8 * hsel + 4 * (l15 & 1);  // lane's first channel; tile t -> cbase+t

    // B_DIM == 2: batch select without an integer division.
    const size_t p = (size_t)point;
    const float* __restrict__ fb =
        (point >= M) ? (feat + (size_t)N * C_DIM) : feat;

    const long long* __restrict__ ip = inds + p * K_DIM;
    const float*     __restrict__ gp = guid + p * (K_DIM * H_DIM);
    const float*     __restrict__ wp = wn   + p * (K_DIM * W_DIM);
    float*           __restrict__ op = out  + p * (C_DIM * W_DIM);

    // ---- Phase 0: neighbor indices -> 8 gathered-row base pointers ----
    const int myidx = (int)ip[l15];         // lanes 0..15 meaningful; N < 2^31
    const float* pA[4];
    const float* pB[4];
#pragma unroll
    for (int s = 0; s < 4; ++s) {
        const int k0 = s * 4 + half * 2;
        const int rA = __shfl(myidx, k0, 32);
        const int rB = __shfl(myidx, k0 + 1, 32);
        pA[s] = fb + (size_t)rA * C_DIM + cbase;   // 16B-aligned float4
        pB[s] = fb + (size_t)rB * C_DIM + cbase;
    }

    // ---- Phase 1: issue ALL loads up front (clauses, imm offsets) ----
    // B operand = weightnet rows (4x16 layout: VGPR0 -> rows k0 / k0+2 per half)
    const float* wpl = wp + half * 2 * W_DIM + l15;
    v2f bm[4];
#pragma unroll
    for (int s = 0; s < 4; ++s) {
        bm[s].x = __builtin_nontemporal_load(wpl + s * 4 * W_DIM);
        bm[s].y = __builtin_nontemporal_load(wpl + s * 4 * W_DIM + W_DIM);
    }

    // Gathered features: one b128 per row; element [t] feeds tile t.
    v4f fA[4], fB[4];
#pragma unroll
    for (int s = 0; s < 4; ++s) {
        fA[s] = *(const v4f*)pA[s];         // 512B per instruction across the wave
        fB[s] = *(const v4f*)pB[s];
    }

    // Guidance: head fixed per lane -> one scalar per K-slot, imm offsets.
    const float* gpl = gp + half * 2 * H_DIM + hsel;
    float gx[4], gy[4];
#pragma unroll
    for (int s = 0; s < 4; ++s) {
        gx[s] = __builtin_nontemporal_load(gpl + s * 4 * H_DIM);
        gy[s] = __builtin_nontemporal_load(gpl + s * 4 * H_DIM + H_DIM);
    }

    // ---- Phase 2: scale + 16 back-to-back WMMAs, store each tile ----
    float* opl = op + half * 32 * W_DIM + l15;   // all store offsets are immediates
#pragma unroll
    for (int t = 0; t < 4; ++t) {
        v8f acc = {0.f, 0.f, 0.f, 0.f, 0.f, 0.f, 0.f, 0.f};
#pragma unroll
        for (int s = 0; s < 4; ++s) {
            v2f a;                          // A: 16x4 f32, lane l15 = row (channel cbase+t)
            a.x = fA[s][t] * gx[s];
            a.y = fB[s][t] * gy[s];
            acc = __builtin_amdgcn_wmma_f32_16x16x4_f32(
                    /*neg_a=*/false, a, /*neg_b=*/false, bm[s],
                    /*c_mod=*/(short)0, acc, /*reuse_a=*/false, /*reuse_b=*/false);
        }
        // D tile: VGPR v -> row M = v (lanes 0-15) / v+8 (lanes 16-31), col = l15.
        // Channel of row M is 8*(M>>1) + 4*(M&1) + t -> compile-time store offsets.
#pragma unroll
        for (int v = 0; v < 8; ++v) {
            const int crel = 8 * (v >> 1) + 4 * (v & 1) + t;   // + 32*half folded into opl
            __builtin_nontemporal_store(acc[v], opl + crel * W_DIM);
        }
    }
}

extern "C" void kernel_launch(void* const* d_in, const int* in_sizes, int n_in,
                              void* d_out, int out_size, void* d_ws, size_t ws_size,
                              hipStream_t stream) {
    const float*     feat = (const float*)d_in[0];       // [B, N, C] fp32
    const long long* inds = (const long long*)d_in[1];   // [B, M, K] int64
    const float*     guid = (const float*)d_in[2];       // [B, M, K, H] fp32
    const float*     wn   = (const float*)d_in[3];       // [B, M, K, C_mid] fp32
    float*           out  = (float*)d_out;               // [B, M, C*C_mid] fp32

    const int N = in_sizes[0] / (B_DIM * C_DIM);
    const int M = in_sizes[1] / (B_DIM * K_DIM);
    const int P = B_DIM * M;

    const int waves_per_block = 256 / 32;                // 8 waves
    const int blocks = (P + waves_per_block - 1) / waves_per_block;
    pcf_wmma_f32_kernel<<<blocks, 256, 0, stream>>>(feat, inds, guid, wn, out, N, M, P);
}